// PMM_13391708029716
// MI455X (gfx1250) — compile-verified
//
#include <hip/hip_runtime.h>
#include <math.h>

typedef __attribute__((ext_vector_type(2))) float v2f;
typedef __attribute__((ext_vector_type(8))) float v8f;

#define Bsz 8
#define Lsz 512
#define Ssz 128
#define Dsz 256
#define NROWS (Bsz * Lsz)          // 4096
#define EPS 1e-12f
#define SHARP 2.0f
#define WTARGET (1.0f / 2000.0f)

// ---------------------------------------------------------------------------
// Kernel 1: Q/K/V = H @ W^T + b via f32 WMMA (16x16x4).
// One wave owns one 16x16 output tile of one of the three matrices.
// A tile (16x4 of H): lanes 0-15 hold M=0..15; vgpr0 = K=0 (lo lanes)/K=2 (hi),
// vgpr1 = K=1/K=3.  B tile (4x16 of W^T): B[k][n] = W[n][k], same row-major
// per-lane addressing as A (row = output col n).
// ---------------------------------------------------------------------------
__global__ void gemm_qkv_kernel(const float* __restrict__ H,
                                const float* __restrict__ Wq, const float* __restrict__ bq,
                                const float* __restrict__ Wk, const float* __restrict__ bk,
                                const float* __restrict__ Wv, const float* __restrict__ bv,
                                float* __restrict__ Q, float* __restrict__ K,
                                float* __restrict__ V) {
    const int gw   = blockIdx.x * (blockDim.x >> 5) + (threadIdx.x >> 5);
    const int lane = threadIdx.x & 31;
    const int mat  = gw >> 12;          // 0..2 : which projection
    const int tile = gw & 4095;         // 256 row-tiles x 16 col-tiles
    const int r0   = (tile >> 4) << 4;  // row base in [0,4096)
    const int n0   = (tile & 15) << 4;  // col base in [0,256)

    const float* W; const float* bias; float* O;
    if (mat == 0)      { W = Wq; bias = bq; O = Q; }
    else if (mat == 1) { W = Wk; bias = bk; O = K; }
    else               { W = Wv; bias = bv; O = V; }

    const int lm   = lane & 15;
    const int koff = (lane >> 4) << 1;  // 0 or 2
    const float* Arow = H + (size_t)(r0 + lm) * Dsz + koff;
    const float* Brow = W + (size_t)(n0 + lm) * Dsz + koff;

    v8f acc = {};
#pragma unroll 4
    for (int k = 0; k < Dsz; k += 4) {
        v2f a = *(const v2f*)(Arow + k);
        v2f b = *(const v2f*)(Brow + k);
        acc = __builtin_amdgcn_wmma_f32_16x16x4_f32(false, a, false, b,
                                                    (short)0, acc, false, false);
    }

    // C/D layout: vgpr i -> row M = i (lanes 0-15) / M = 8+i (lanes 16-31)
    const int col   = n0 + lm;
    const int rbase = r0 + ((lane >> 4) << 3);
    const float bs  = bias[col];
#pragma unroll
    for (int i = 0; i < 8; ++i)
        O[(size_t)(rbase + i) * Dsz + col] = acc[i] + bs;
}

// ---------------------------------------------------------------------------
// Kernel 2: L2-normalize Q,K rows in place; gate = sigmoid(H . Wg + bg).
// One wave per row; shuffle reductions (wave32).
// ---------------------------------------------------------------------------
__global__ void norm_gate_kernel(const float* __restrict__ H,
                                 const float* __restrict__ Wg, const float* __restrict__ bg,
                                 float* __restrict__ Q, float* __restrict__ K,
                                 float* __restrict__ G) {
    const int r    = blockIdx.x * (blockDim.x >> 5) + (threadIdx.x >> 5);
    const int lane = threadIdx.x & 31;
    float* qr = Q + (size_t)r * Dsz;
    float* kr = K + (size_t)r * Dsz;
    const float* hr = H + (size_t)r * Dsz;

    float qv[8], kv[8];
    float sq = 0.f, sk = 0.f, sg = 0.f;
#pragma unroll
    for (int i = 0; i < 8; ++i) {
        const int d = lane + 32 * i;
        qv[i] = qr[d]; kv[i] = kr[d];
        sq += qv[i] * qv[i];
        sk += kv[i] * kv[i];
        sg += hr[d] * Wg[d];
    }
    for (int off = 16; off; off >>= 1) {
        sq += __shfl_xor(sq, off);
        sk += __shfl_xor(sk, off);
        sg += __shfl_xor(sg, off);
    }
    const float iq = 1.f / fmaxf(sqrtf(sq), EPS);
    const float ik = 1.f / fmaxf(sqrtf(sk), EPS);
#pragma unroll
    for (int i = 0; i < 8; ++i) {
        const int d = lane + 32 * i;
        qr[d] = qv[i] * iq;
        kr[d] = kv[i] * ik;
    }
    if (lane == 0)
        G[r] = 1.f / (1.f + __expf(-(sg + bg[0])));
}

// ---------------------------------------------------------------------------
// Kernel 3: sp = mean_t (mean_b gate - target)^2 -> out[B*L*D]
// ---------------------------------------------------------------------------
__global__ void sp_kernel(const float* __restrict__ G, float* __restrict__ out_sp) {
    __shared__ float red[Lsz];
    const int t = threadIdx.x;
    float s = 0.f;
#pragma unroll
    for (int b = 0; b < Bsz; ++b) s += G[b * Lsz + t];
    const float dv = s * (1.0f / Bsz) - WTARGET;
    red[t] = dv * dv;
    __syncthreads();
    for (int off = Lsz >> 1; off; off >>= 1) {
        if (t < off) red[t] += red[t + off];
        __syncthreads();
    }
    if (t == 0) out_sp[0] = red[0] * (1.0f / Lsz);
}

// ---------------------------------------------------------------------------
// Kernel 4: the sequential scan. One workgroup per batch; mem (128KB) lives
// in LDS for all 512 steps (CDNA5 WGP LDS = 320KB). 512 threads = 16 waves.
// Thread mapping for dot/update passes: s = tid>>2 (slot), quarter = tid&3
// (64 contiguous dims). Quad reductions via shfl_xor(1),(2).
// ---------------------------------------------------------------------------
__global__ void scan_kernel(const float* __restrict__ initm,
                            const float* __restrict__ masks,
                            const float* __restrict__ Qn,
                            const float* __restrict__ Kn,
                            const float* __restrict__ V,
                            const float* __restrict__ G,
                            float* __restrict__ out) {
    extern __shared__ float smf[];
    float* mem   = smf;                    // 128*256
    float* qs    = mem + Ssz * Dsz;        // 256
    float* ks    = qs + Dsz;               // 256
    float* vs    = ks + Dsz;               // 256
    float* simb  = vs + Dsz;               // 128
    float* wsimb = simb + Ssz;             // 128
    float* rw    = wsimb + Ssz;            // 128  read weights
    float* wc    = rw + Ssz;               // 128  gate*write weights
    float* denom = wc + Ssz;               // 128  max(||mem_s||, eps)
    float* rvp   = denom + Ssz;            // 512  read_v partials

    const int tid  = threadIdx.x;
    const int b    = blockIdx.x;
    const int lane = tid & 31;
    const int s    = tid >> 2;
    const int q4   = tid & 3;
    const int d0   = q4 * 64;
    float* mrow = mem + s * Dsz + d0;

    // ---- init memory + slot norms ----
    const float* im = initm + (size_t)b * Ssz * Dsz + s * Dsz + d0;
    float ssq = 0.f;
#pragma unroll 8
    for (int j = 0; j < 64; ++j) {
        const float v = im[j];
        mrow[j] = v;
        ssq += v * v;
    }
    ssq += __shfl_xor(ssq, 1);
    ssq += __shfl_xor(ssq, 2);
    if (q4 == 0) denom[s] = fmaxf(sqrtf(ssq), EPS);
    __syncthreads();

    for (int t = 0; t < Lsz; ++t) {
        const int r = b * Lsz + t;
        // stage q_n, k_n, wv for this step
        if (tid < Dsz) {
            qs[tid] = Qn[(size_t)r * Dsz + tid];
            vs[tid] = V[(size_t)r * Dsz + tid];
        } else {
            const int u = tid - Dsz;
            ks[u] = Kn[(size_t)r * Dsz + u];
        }
        __syncthreads();

        // ---- cosine sims against all slots ----
        float pq = 0.f, pk = 0.f;
#pragma unroll 8
        for (int j = 0; j < 64; ++j) {
            const float mv = mrow[j];
            pq += mv * qs[d0 + j];
            pk += mv * ks[d0 + j];
        }
        pq += __shfl_xor(pq, 1); pq += __shfl_xor(pq, 2);
        pk += __shfl_xor(pk, 1); pk += __shfl_xor(pk, 2);
        if (q4 == 0) {
            const float inv = 1.f / denom[s];
            simb[s]  = pq * inv;
            wsimb[s] = pk * inv;
        }
        __syncthreads();

        // ---- two softmaxes over S=128, wave0 (read) & wave1 (write) in parallel
        const int wid = tid >> 5;
        if (wid == 0) {
            float x0 = SHARP * simb[lane],      x1 = SHARP * simb[lane + 32];
            float x2 = SHARP * simb[lane + 64], x3 = SHARP * simb[lane + 96];
            float mx = fmaxf(fmaxf(x0, x1), fmaxf(x2, x3));
            for (int off = 16; off; off >>= 1) mx = fmaxf(mx, __shfl_xor(mx, off));
            float e0 = __expf(x0 - mx), e1 = __expf(x1 - mx);
            float e2 = __expf(x2 - mx), e3 = __expf(x3 - mx);
            float sme = e0 + e1 + e2 + e3;
            for (int off = 16; off; off >>= 1) sme += __shfl_xor(sme, off);
            const float inv = 1.f / sme;
            rw[lane] = e0 * inv; rw[lane + 32] = e1 * inv;
            rw[lane + 64] = e2 * inv; rw[lane + 96] = e3 * inv;
        } else if (wid == 1) {
            const float g = G[r];
            float x0 = SHARP * wsimb[lane],      x1 = SHARP * wsimb[lane + 32];
            float x2 = SHARP * wsimb[lane + 64], x3 = SHARP * wsimb[lane + 96];
            float mx = fmaxf(fmaxf(x0, x1), fmaxf(x2, x3));
            for (int off = 16; off; off >>= 1) mx = fmaxf(mx, __shfl_xor(mx, off));
            float e0 = __expf(x0 - mx), e1 = __expf(x1 - mx);
            float e2 = __expf(x2 - mx), e3 = __expf(x3 - mx);
            float sme = e0 + e1 + e2 + e3;
            for (int off = 16; off; off >>= 1) sme += __shfl_xor(sme, off);
            const float gi = g / sme;
            wc[lane] = e0 * gi; wc[lane + 32] = e1 * gi;
            wc[lane + 64] = e2 * gi; wc[lane + 96] = e3 * gi;
        }
        __syncthreads();

        // ---- read_v partials: d = tid&255, half of slots = tid>>8 ----
        {
            const int d = tid & 255, hh = tid >> 8;
            const int sb = hh * 64;
            float acc = 0.f;
#pragma unroll 8
            for (int j = 0; j < 64; ++j)
                acc += rw[sb + j] * mem[(sb + j) * Dsz + d];
            rvp[tid] = acc;
        }
        __syncthreads();

        if (tid < Dsz)
            out[(size_t)r * Dsz + tid] = rvp[tid] + rvp[tid + Dsz];

        // ---- gated write + mask + next-step norms ----
        const float mk = masks[r];
        const float w  = wc[s];
        float ssq2 = 0.f;
#pragma unroll 8
        for (int j = 0; j < 64; ++j) {
            const float mv = mrow[j];
            const float nv = (mv + w * (vs[d0 + j] - mv)) * mk;
            mrow[j] = nv;
            ssq2 += nv * nv;
        }
        ssq2 += __shfl_xor(ssq2, 1);
        ssq2 += __shfl_xor(ssq2, 2);
        if (q4 == 0) denom[s] = fmaxf(sqrtf(ssq2), EPS);
        __syncthreads();
    }
}

// ---------------------------------------------------------------------------
extern "C" void kernel_launch(void* const* d_in, const int* in_sizes, int n_in,
                              void* d_out, int out_size, void* d_ws, size_t ws_size,
                              hipStream_t stream) {
    const float* initm = (const float*)d_in[0];
    const float* H     = (const float*)d_in[1];
    const float* masks = (const float*)d_in[2];
    const float* Wq = (const float*)d_in[3];  const float* bq = (const float*)d_in[4];
    const float* Wk = (const float*)d_in[5];  const float* bk = (const float*)d_in[6];
    const float* Wv = (const float*)d_in[7];  const float* bv = (const float*)d_in[8];
    const float* Wg = (const float*)d_in[9];  const float* bg = (const float*)d_in[10];
    float* out = (float*)d_out;

    // workspace: Q,K,V (4096x256 each) + G (4096)  ~= 12.6 MB
    float* Q = (float*)d_ws;
    float* K = Q + (size_t)NROWS * Dsz;
    float* V = K + (size_t)NROWS * Dsz;
    float* G = V + (size_t)NROWS * Dsz;

    // K1: 3 * 4096 tiles, 8 waves (256 thr) per block -> 1536 blocks
    gemm_qkv_kernel<<<1536, 256, 0, stream>>>(H, Wq, bq, Wk, bk, Wv, bv, Q, K, V);
    // K2: 4096 rows, wave per row, 8 waves per block -> 512 blocks
    norm_gate_kernel<<<512, 256, 0, stream>>>(H, Wg, bg, Q, K, G);
    // K3: sp scalar -> out[B*L*D]
    sp_kernel<<<1, Lsz, 0, stream>>>(G, out + (size_t)Bsz * Lsz * Dsz);
    // K4: scan; dynamic LDS = mem + staging + softmax scratch
    const size_t lds = (size_t)(Ssz * Dsz + 3 * Dsz + 5 * Ssz + 512) * sizeof(float);
    scan_kernel<<<Bsz, 512, lds, stream>>>(initm, masks, Q, K, V, G, out);
}